// STPAttention_Bad_15891378995684
// MI455X (gfx1250) — compile-verified
//
#include <hip/hip_runtime.h>
#include <hip/hip_bf16.h>

// ---------------------------------------------------------------------------
// STP attention (head_dim = 3) for MI455X / gfx1250.
//   D=384, H=128, HD=3, B=2, T=1024, M = B*T = 2048.
//   q = x Wq^T + bq ; k = x Wk^T + bk ; v = x Wv^T + bv
//   kmod = k + cross(k, v)
//   ctx  = causal-softmax(q . kmod / sqrt(3)) @ v      (flash-style)
//   out  = ctx Wo^T + bo
// All matrix math on V_WMMA_F32_16X16X4_F32 (fp32 WMMA, one 16x16 tile/wave).
// ---------------------------------------------------------------------------

#define D_MODEL   384
#define NUM_HEADS 128
#define SEQ_T     1024
#define BATCH     2
#define M_ROWS    (BATCH * SEQ_T)          // 2048
#define STP_SCALE 0.57735026918962576451f  // 1/sqrt(3)

typedef __attribute__((ext_vector_type(2))) float v2f;
typedef __attribute__((ext_vector_type(8))) float v8f;

// ---------------------------------------------------------------------------
// Kernel 1/4: Out[M,N] = X[M,K] * W[N,K]^T + bias[N]   (fp32 WMMA 16x16x4)
// One wave = one 16x16 output tile; 4 waves/block cover a 16x64 strip.
// A frag: lane L holds A[m = L&15][k = 2*(L>>4)+r], r=0..1  -> float2 load
// B frag: lane L holds B[k = 2*(L>>4)+r][n = L&15] = W[n][k] -> float2 load
// D frag: lane L, vgpr r holds D[m = r + 8*(L>>4)][n = L&15]
// ---------------------------------------------------------------------------
__global__ __launch_bounds__(128) void gemm_xwT_bias(
    const float* __restrict__ X, const float* __restrict__ W,
    const float* __restrict__ bias, float* __restrict__ Out,
    int M, int N, int K) {
  const int lane  = threadIdx.x & 31;
  const int wave  = threadIdx.x >> 5;
  const int row0  = blockIdx.x * 16;
  const int col0  = blockIdx.y * 64 + wave * 16;
  const int mn    = lane & 15;         // A-row / B-col carried by this lane
  const int koff  = (lane >> 4) * 2;   // K offset within a K=4 step: {0,2}

  const float* aptr = X + (size_t)(row0 + mn) * K + koff;
  const float* bptr = W + (size_t)(col0 + mn) * K + koff;  // W is [N,K] row-major

  v8f acc = {};
  for (int kk = 0; kk < K; kk += 4) {
    v2f a = *(const v2f*)(aptr + kk);
    v2f b = *(const v2f*)(bptr + kk);
    acc = __builtin_amdgcn_wmma_f32_16x16x4_f32(false, a, false, b,
                                                (short)0, acc, false, false);
  }

  const float bn = bias[col0 + mn];    // this lane owns one output column
  float* orow = Out + (size_t)(row0 + 8 * (lane >> 4)) * N + col0 + mn;
#pragma unroll
  for (int r = 0; r < 8; ++r)
    orow[(size_t)r * N] = acc[r] + bn;
}

// ---------------------------------------------------------------------------
// Kernel 2: kmod = k + cross(k, v), per (row, head) group of 3 columns.
// ---------------------------------------------------------------------------
__global__ __launch_bounds__(256) void cross_fuse(
    const float* __restrict__ K, const float* __restrict__ V,
    float* __restrict__ Kmod, int total) {
  int i = blockIdx.x * blockDim.x + threadIdx.x;
  if (i >= total) return;                          // total = M_ROWS * NUM_HEADS
  int row = i >> 7;                                // / NUM_HEADS
  int h   = i & (NUM_HEADS - 1);
  size_t base = (size_t)row * D_MODEL + h * 3;
  float k0 = K[base], k1 = K[base + 1], k2 = K[base + 2];
  float w0 = V[base], w1 = V[base + 1], w2 = V[base + 2];
  Kmod[base + 0] = k0 + (k1 * w2 - k2 * w1);
  Kmod[base + 1] = k1 + (k2 * w0 - k0 * w2);
  Kmod[base + 2] = k2 + (k0 * w1 - k1 * w0);
}

// ---------------------------------------------------------------------------
// Kernel 3: flash attention, one wave per 16-query tile of one (b,h) slice.
// Score tile 16x16 = ONE f32 WMMA (K=4: head_dim 3 padded with zero).
// Online softmax: row-max reduced across the 16 lanes of a row each tile;
// l and the P@V accumulator kept as per-lane partials, reduced once at end
// (valid because alpha = exp(m_old-m_new) is identical across a row's lanes).
// ---------------------------------------------------------------------------
__global__ __launch_bounds__(128) void stp_flash_attn(
    const float* __restrict__ Q, const float* __restrict__ Kmod,
    const float* __restrict__ V, float* __restrict__ Ctx) {
  const int lane  = threadIdx.x & 31;
  const int wave  = threadIdx.x >> 5;
  const int qTile = blockIdx.x * 4 + wave;          // 0..63
  const int bh    = blockIdx.y;                     // 0..255
  const int b     = bh >> 7;
  const int h     = bh & (NUM_HEADS - 1);
  const int qBase = qTile * 16;
  const int n     = lane & 15;                      // key-in-tile / out column
  const int half  = lane >> 4;                      // 0|1
  const int d0    = 2 * half;                       // head-dim slot {0,2}

  // Q fragment (A matrix, 16 queries x K=4 padded head dims)
  v2f aq;
  {
    const float* qrow = Q + ((size_t)(b * SEQ_T + qBase + n)) * D_MODEL + h * 3;
    aq.x = qrow[d0];
    aq.y = (d0 + 1 < 3) ? qrow[d0 + 1] : 0.0f;
  }

  float mrun[8], lrun[8], a0[8], a1[8], a2[8];
#pragma unroll
  for (int r = 0; r < 8; ++r) {
    mrun[r] = -1e30f; lrun[r] = 0.0f;
    a0[r] = 0.0f; a1[r] = 0.0f; a2[r] = 0.0f;
  }

  const int nTiles = qTile + 1;                     // causal: keys <= qBase+15
  for (int kt = 0; kt < nTiles; ++kt) {
    const int kBase = kt * 16;
    // Kmod fragment (B matrix, K=4 padded head dims x 16 keys)
    const float* krow =
        Kmod + ((size_t)(b * SEQ_T + kBase + n)) * D_MODEL + h * 3;
    v2f bk;
    bk.x = krow[d0];
    bk.y = (d0 + 1 < 3) ? krow[d0 + 1] : 0.0f;

    v8f s = {};
    s = __builtin_amdgcn_wmma_f32_16x16x4_f32(false, aq, false, bk,
                                              (short)0, s, false, false);

    // V row for this lane's key (same key for both wave halves)
    const float* vrow =
        V + ((size_t)(b * SEQ_T + kBase + n)) * D_MODEL + h * 3;
    const float v0 = vrow[0], v1 = vrow[1], v2 = vrow[2];
    const int ki = kBase + n;

#pragma unroll
    for (int r = 0; r < 8; ++r) {
      const int qi = qBase + r + 8 * half;          // row owned by vgpr r
      float sv = (ki <= qi) ? s[r] * STP_SCALE : -1e30f;
      // row max across the 16 lanes holding this row (stays in half-wave)
      float mt = sv;
      mt = fmaxf(mt, __shfl_xor(mt, 1));
      mt = fmaxf(mt, __shfl_xor(mt, 2));
      mt = fmaxf(mt, __shfl_xor(mt, 4));
      mt = fmaxf(mt, __shfl_xor(mt, 8));
      const float mnew  = fmaxf(mrun[r], mt);
      const float alpha = __expf(mrun[r] - mnew);   // same on all 16 lanes
      const float p     = __expf(sv - mnew);
      lrun[r] = lrun[r] * alpha + p;                // per-lane partials
      a0[r]   = a0[r]   * alpha + p * v0;
      a1[r]   = a1[r]   * alpha + p * v1;
      a2[r]   = a2[r]   * alpha + p * v2;
      mrun[r] = mnew;
    }
  }

  // Final 16-lane reductions + normalize + store ctx[row][h*3 + c], c = n < 3.
#pragma unroll
  for (int r = 0; r < 8; ++r) {
    float l = lrun[r], c0 = a0[r], c1 = a1[r], c2 = a2[r];
#pragma unroll
    for (int mask = 1; mask <= 8; mask <<= 1) {
      l  += __shfl_xor(l,  mask);
      c0 += __shfl_xor(c0, mask);
      c1 += __shfl_xor(c1, mask);
      c2 += __shfl_xor(c2, mask);
    }
    const float inv = 1.0f / l;
    const int row = qBase + r + 8 * half;
    float* crow = Ctx + ((size_t)(b * SEQ_T + row)) * D_MODEL + h * 3;
    float outv = (n == 0) ? c0 * inv : (n == 1) ? c1 * inv : c2 * inv;
    if (n < 3) crow[n] = outv;
  }
}

// ---------------------------------------------------------------------------
// Launch: q|k|v projections -> cross fuse -> flash attn -> output projection.
// Workspace: q, k, v, kmod (3 MB each); ctx reuses the dead k buffer.
// ---------------------------------------------------------------------------
extern "C" void kernel_launch(void* const* d_in, const int* in_sizes, int n_in,
                              void* d_out, int out_size, void* d_ws,
                              size_t ws_size, hipStream_t stream) {
  const float* x  = (const float*)d_in[0];
  const float* Wq = (const float*)d_in[1];
  const float* bq = (const float*)d_in[2];
  const float* Wk = (const float*)d_in[3];
  const float* bk = (const float*)d_in[4];
  const float* Wv = (const float*)d_in[5];
  const float* bv = (const float*)d_in[6];
  const float* Wo = (const float*)d_in[7];
  const float* bo = (const float*)d_in[8];
  float* out = (float*)d_out;

  const size_t mat = (size_t)M_ROWS * D_MODEL;      // 786432 floats
  float* q    = (float*)d_ws;
  float* k    = q + mat;
  float* v    = k + mat;
  float* kmod = v + mat;
  float* ctx  = k;                                  // k dead after cross_fuse

  dim3 gGemm(M_ROWS / 16, D_MODEL / 64);            // 128 x 6
  gemm_xwT_bias<<<gGemm, 128, 0, stream>>>(x, Wq, bq, q, M_ROWS, D_MODEL, D_MODEL);
  gemm_xwT_bias<<<gGemm, 128, 0, stream>>>(x, Wk, bk, k, M_ROWS, D_MODEL, D_MODEL);
  gemm_xwT_bias<<<gGemm, 128, 0, stream>>>(x, Wv, bv, v, M_ROWS, D_MODEL, D_MODEL);

  const int total = M_ROWS * NUM_HEADS;
  cross_fuse<<<(total + 255) / 256, 256, 0, stream>>>(k, v, kmod, total);

  dim3 gAttn((SEQ_T / 16) / 4, BATCH * NUM_HEADS);  // 16 x 256
  stp_flash_attn<<<gAttn, 128, 0, stream>>>(q, kmod, v, ctx);

  gemm_xwT_bias<<<gGemm, 128, 0, stream>>>(ctx, Wo, bo, out, M_ROWS, D_MODEL, D_MODEL);
}